// ClusterGCN_75625784148632
// MI455X (gfx1250) — compile-verified
//
#include <hip/hip_runtime.h>

#define N_NODES  50000
#define N_EDGES  1600000
#define HID      128
#define OUT_CH   2
#define N_GRAPHS 512
#define N_HID_LAYERS 5
#define EPS 1e-5f

typedef float v2f __attribute__((ext_vector_type(2)));
typedef float v8f __attribute__((ext_vector_type(8)));

__global__ void fill_kernel(float* __restrict__ p, int n, float v) {
    int i = blockIdx.x * blockDim.x + threadIdx.x;
    if (i < n) p[i] = v;
}

__global__ void degree_kernel(const int* __restrict__ dst, float* __restrict__ deg) {
    int e = blockIdx.x * blockDim.x + threadIdx.x;
    if (e < N_EDGES) atomicAdd(&deg[dst[e]], 1.0f);
}

__global__ void invdeg_kernel(float* __restrict__ deg) {
    int i = blockIdx.x * blockDim.x + threadIdx.x;
    if (i < N_NODES) deg[i] = 1.0f / fmaxf(deg[i], 1.0f);
}

// One wave per edge; each lane moves 4 channels (float4 gather + 4 atomic adds).
__global__ void __launch_bounds__(256)
aggregate_kernel(const float* __restrict__ h,
                 const int* __restrict__ src,
                 const int* __restrict__ dst,
                 float* __restrict__ agg) {
    long long gid = (long long)blockIdx.x * blockDim.x + threadIdx.x;
    int e = (int)(gid >> 5);
    if (e >= N_EDGES) return;
    int cg = ((int)gid & 31) * 4;
    int s = src[e];
    int d = dst[e];
    const float4 v = *(const float4*)(h + (long long)s * HID + cg);
    float* ap = agg + (long long)d * HID + cg;
    atomicAdd(ap + 0, v.x);
    atomicAdd(ap + 1, v.y);
    atomicAdd(ap + 2, v.z);
    atomicAdd(ap + 3, v.w);
}

// Fused: out = relu(BN(agg*inv_deg @ Wl + h @ Wr + bias))
// Block = 16 nodes x 128 cols; 8 waves, each wave owns one 16x16 N-tile.
// Uses V_WMMA_F32_16X16X4_F32 (full fp32), 64 WMMAs per wave.
__global__ void __launch_bounds__(256)
sage_gemm_bn_relu(const float* __restrict__ agg,
                  const float* __restrict__ h,
                  const float* __restrict__ inv_deg,
                  const float* __restrict__ Wl,
                  const float* __restrict__ Wr,
                  const float* __restrict__ bias,
                  const float* __restrict__ gamma,
                  const float* __restrict__ beta,
                  const float* __restrict__ rmean,
                  const float* __restrict__ rvar,
                  float* __restrict__ out) {
    const int lane  = threadIdx.x & 31;
    const int wv    = threadIdx.x >> 5;      // N-tile index 0..7
    const int l16   = lane & 15;
    const int khalf = lane >> 4;             // 0 or 1
    const int row0  = blockIdx.x * 16;       // 3125 * 16 == 50000, exact
    const int arow  = row0 + l16;            // A-operand row for this lane
    const int n     = wv * 16 + l16;         // output column (0..127)

    const float invd = inv_deg[arow];
    const float* __restrict__ ap = agg + (long long)arow * HID;
    const float* __restrict__ hp = h   + (long long)arow * HID;

    v8f c = {};
    #pragma unroll 4
    for (int kk = 0; kk < HID; kk += 4) {
        const int k = kk + khalf * 2;
        v2f a, bm;
        // neighbor-mean path: (agg * inv_deg) @ Wl
        a.x  = ap[k]     * invd;
        a.y  = ap[k + 1] * invd;
        bm.x = Wl[k * HID + n];
        bm.y = Wl[(k + 1) * HID + n];
        c = __builtin_amdgcn_wmma_f32_16x16x4_f32(false, a, false, bm,
                                                  (short)0, c, false, false);
        // root path: h @ Wr
        a.x  = hp[k];
        a.y  = hp[k + 1];
        bm.x = Wr[k * HID + n];
        bm.y = Wr[(k + 1) * HID + n];
        c = __builtin_amdgcn_wmma_f32_16x16x4_f32(false, a, false, bm,
                                                  (short)0, c, false, false);
    }

    // BN(eval) folded: val = (acc + bias - rmean)*scale + beta, then ReLU
    const float scale = gamma[n] * rsqrtf(rvar[n] + EPS);
    const float shift = (bias[n] - rmean[n]) * scale + beta[n];
    #pragma unroll
    for (int v = 0; v < 8; ++v) {
        const int mm = row0 + v + 8 * khalf;   // D row per C/D VGPR layout
        float val = fmaxf(c[v] * scale + shift, 0.0f);
        out[(long long)mm * HID + n] = val;
    }
}

// Final SAGEConv 128->2 fused with global_mean_pool accumulation.
__global__ void __launch_bounds__(256)
final_conv_pool(const float* __restrict__ agg,
                const float* __restrict__ h,
                const float* __restrict__ inv_deg,
                const float* __restrict__ Wl,   // [128][2]
                const float* __restrict__ Wr,   // [128][2]
                const float* __restrict__ bo,   // [2]
                const int* __restrict__ batch,
                float* __restrict__ gsum,       // [512*2]
                float* __restrict__ gcnt) {     // [512]
    int node = blockIdx.x * blockDim.x + threadIdx.x;
    if (node >= N_NODES) return;
    const float invd = inv_deg[node];
    const float* __restrict__ ap = agg + (long long)node * HID;
    const float* __restrict__ hp = h   + (long long)node * HID;
    float acc0 = bo[0], acc1 = bo[1];
    #pragma unroll 4
    for (int k = 0; k < HID; ++k) {
        float av = ap[k] * invd;
        float hv = hp[k];
        acc0 += av * Wl[k * 2 + 0] + hv * Wr[k * 2 + 0];
        acc1 += av * Wl[k * 2 + 1] + hv * Wr[k * 2 + 1];
    }
    int g = batch[node];
    atomicAdd(&gsum[g * 2 + 0], acc0);
    atomicAdd(&gsum[g * 2 + 1], acc1);
    atomicAdd(&gcnt[g], 1.0f);
}

__global__ void pool_div(const float* __restrict__ gsum,
                         const float* __restrict__ gcnt,
                         float* __restrict__ out) {
    int i = blockIdx.x * blockDim.x + threadIdx.x;
    if (i < N_GRAPHS * OUT_CH)
        out[i] = gsum[i] / fmaxf(gcnt[i >> 1], 1.0f);
}

extern "C" void kernel_launch(void* const* d_in, const int* in_sizes, int n_in,
                              void* d_out, int out_size, void* d_ws, size_t ws_size,
                              hipStream_t stream) {
    const float* x      = (const float*)d_in[0];
    const float* Wl     = (const float*)d_in[1];   // [5,128,128]
    const float* Wr     = (const float*)d_in[2];   // [5,128,128]
    const float* b      = (const float*)d_in[3];   // [5,128]
    const float* gamma  = (const float*)d_in[4];
    const float* beta   = (const float*)d_in[5];
    const float* rmean  = (const float*)d_in[6];
    const float* rvar   = (const float*)d_in[7];
    const float* Wl_out = (const float*)d_in[8];   // [128,2]
    const float* Wr_out = (const float*)d_in[9];   // [128,2]
    const float* b_out  = (const float*)d_in[10];  // [2]
    const int*   edge   = (const int*)d_in[11];    // [2,1.6M]
    const int*   batch  = (const int*)d_in[12];    // [50000]
    const int* src = edge;
    const int* dst = edge + N_EDGES;

    float* ws = (float*)d_ws;
    const size_t NF = (size_t)N_NODES * HID;       // 6.4M floats per feature buffer
    float* buf0 = ws;
    float* buf1 = buf0 + NF;
    float* agg  = buf1 + NF;
    float* ideg = agg + NF;                        // [50000]
    float* gsum = ideg + N_NODES;                  // [1024]
    float* gcnt = gsum + (size_t)N_GRAPHS * OUT_CH;// [512]

    // degree -> inv_deg (once; shared by all layers)
    fill_kernel<<<(N_NODES + 255) / 256, 256, 0, stream>>>(ideg, N_NODES, 0.0f);
    degree_kernel<<<(N_EDGES + 255) / 256, 256, 0, stream>>>(dst, ideg);
    invdeg_kernel<<<(N_NODES + 255) / 256, 256, 0, stream>>>(ideg);

    const int aggBlocks = (int)(((long long)N_EDGES * 32 + 255) / 256);  // 200000
    const int nfBlocks  = (int)((NF + 255) / 256);

    const float* hin = x;
    float* hout = buf0;
    for (int l = 0; l < N_HID_LAYERS; ++l) {
        fill_kernel<<<nfBlocks, 256, 0, stream>>>(agg, (int)NF, 0.0f);
        aggregate_kernel<<<aggBlocks, 256, 0, stream>>>(hin, src, dst, agg);
        sage_gemm_bn_relu<<<N_NODES / 16, 256, 0, stream>>>(
            agg, hin, ideg,
            Wl + (size_t)l * HID * HID, Wr + (size_t)l * HID * HID, b + l * HID,
            gamma + l * HID, beta + l * HID, rmean + l * HID, rvar + l * HID,
            hout);
        hin  = hout;
        hout = (hout == buf0) ? buf1 : buf0;
    }

    // final conv (128->2) + global mean pool
    fill_kernel<<<nfBlocks, 256, 0, stream>>>(agg, (int)NF, 0.0f);
    aggregate_kernel<<<aggBlocks, 256, 0, stream>>>(hin, src, dst, agg);
    fill_kernel<<<(N_GRAPHS * OUT_CH + N_GRAPHS + 255) / 256, 256, 0, stream>>>(
        gsum, N_GRAPHS * OUT_CH + N_GRAPHS, 0.0f);
    final_conv_pool<<<(N_NODES + 255) / 256, 256, 0, stream>>>(
        agg, hin, ideg, Wl_out, Wr_out, b_out, batch, gsum, gcnt);
    pool_div<<<(N_GRAPHS * OUT_CH + 255) / 256, 256, 0, stream>>>(
        gsum, gcnt, (float*)d_out);
}